// GCN_38362647888479
// MI455X (gfx1250) — compile-verified
//
#include <hip/hip_runtime.h>

typedef float v2f __attribute__((ext_vector_type(2)));
typedef float v8f __attribute__((ext_vector_type(8)));

// ---------------------------------------------------------------------------
// init: zero the aggregation target (h region of d_out) and set deg=1 (self
// loop contribution to degree).
// ---------------------------------------------------------------------------
__global__ void __launch_bounds__(256) init_kernel(float* __restrict__ h,
                                                   float* __restrict__ deg,
                                                   int n_nodes) {
  long long i = (long long)blockIdx.x * blockDim.x + threadIdx.x;
  long long total = (long long)n_nodes * 128;
  if (i < total) h[i] = 0.0f;
  if (i < n_nodes) deg[i] = 1.0f;
}

// ---------------------------------------------------------------------------
// degree count on destination nodes
// ---------------------------------------------------------------------------
__global__ void __launch_bounds__(256) deg_kernel(const int* __restrict__ dst,
                                                  float* __restrict__ deg, int E) {
  int e = blockIdx.x * blockDim.x + threadIdx.x;
  if (e < E) atomicAdd(&deg[dst[e]], 1.0f);
}

// deg -> rsqrt(deg) in place (deg >= 1 thanks to self loops)
__global__ void __launch_bounds__(256) rsqrt_kernel(float* __restrict__ deg, int n) {
  int i = blockIdx.x * blockDim.x + threadIdx.x;
  if (i < n) deg[i] = rsqrtf(deg[i]);
}

// ---------------------------------------------------------------------------
// GEMM1: XW = X[ n x 256 ] * W[ 256 x 128 ]   (fp32 WMMA 16x16x4)
// One block = 8 waves = one 16-row strip; wave w computes 16x16 tile at
// columns [16w, 16w+16). A strip staged in LDS once, shared by all 8 waves.
// ---------------------------------------------------------------------------
__global__ void __launch_bounds__(256) gemm1_wmma(const float* __restrict__ X,
                                                  const float* __restrict__ W,
                                                  float* __restrict__ XW,
                                                  int n_nodes) {
  __shared__ float a_tile[16 * 260];  // pad 256->260 to avoid LDS bank conflicts
  const int tid = threadIdx.x;
  const int rowBase = blockIdx.x * 16;

  // stage 16x256 A strip (coalesced: 16 full-width rows)
  for (int idx = tid; idx < 16 * 256; idx += 256) {
    int r = idx >> 8, c = idx & 255;
    a_tile[r * 260 + c] = X[(long long)(rowBase + r) * 256 + c];
  }
  __syncthreads();

  const int wave = tid >> 5;
  const int lane = tid & 31;
  const int half = lane >> 4;  // 0: K pair {0,1}, 1: K pair {2,3}
  const int lrow = lane & 15;
  const int colBase = wave * 16;

  v8f acc = {};
  for (int k0 = 0; k0 < 256; k0 += 4) {
    const int ka = k0 + 2 * half;
    v2f a, b;
    // 32-bit A 16x4 layout: lanes 0-15 K=0,1 ; lanes 16-31 K=2,3 (per row M=lrow)
    a[0] = a_tile[lrow * 260 + ka];
    a[1] = a_tile[lrow * 260 + ka + 1];
    // B 4x16: row K striped across lanes (N = lrow), halves hold K pairs
    b[0] = W[ka * 128 + colBase + lrow];
    b[1] = W[(ka + 1) * 128 + colBase + lrow];
    acc = __builtin_amdgcn_wmma_f32_16x16x4_f32(false, a, false, b,
                                                (short)0, acc, false, false);
  }

  // 32-bit C/D layout: VGPR v -> M = v + 8*half, N = lrow
#pragma unroll
  for (int v = 0; v < 8; ++v) {
    XW[(long long)(rowBase + v + 8 * half) * 128 + colBase + lrow] = acc[v];
  }
}

// ---------------------------------------------------------------------------
// edge scatter: one wave per edge; lane gathers float4 of xw[src], scales by
// dinv[src]*dinv[dst], atomically accumulates into agg[dst]. xw + agg are
// L2-resident (51 MB each vs 192 MB L2) so atomics RMW in L2.
// ---------------------------------------------------------------------------
__global__ void __launch_bounds__(256) scatter_kernel(const int* __restrict__ ei,
                                                      const float* __restrict__ xw,
                                                      const float* __restrict__ dinv,
                                                      float* __restrict__ agg,
                                                      int E) {
  int w = (int)(((long long)blockIdx.x * blockDim.x + threadIdx.x) >> 5);
  int lane = threadIdx.x & 31;
  if (w >= E) return;
  int s = ei[w];       // edge_index[0][w]
  int d = ei[E + w];   // edge_index[1][w]
  float nw = dinv[s] * dinv[d];
  const float4* xs = (const float4*)(xw + (long long)s * 128);
  float4 v = xs[lane];
  float* dp = agg + (long long)d * 128 + lane * 4;
  atomicAdd(dp + 0, v.x * nw);
  atomicAdd(dp + 1, v.y * nw);
  atomicAdd(dp + 2, v.z * nw);
  atomicAdd(dp + 3, v.w * nw);
}

// ---------------------------------------------------------------------------
// finalize: h = relu(agg + xw*dinv^2 (self loop) + b_gcn), float4-vectorized,
// in place on the h region of d_out.
// ---------------------------------------------------------------------------
__global__ void __launch_bounds__(256) finalize_kernel(float* __restrict__ h,
                                                       const float* __restrict__ xw,
                                                       const float* __restrict__ dinv,
                                                       const float* __restrict__ b,
                                                       int n_nodes) {
  long long i = (long long)blockIdx.x * blockDim.x + threadIdx.x;  // float4 index
  long long total = (long long)n_nodes * 32;
  if (i >= total) return;
  int row = (int)(i >> 5);
  int f4 = (int)(i & 31);
  float di = dinv[row];
  float wsl = di * di;
  float4 hv = ((float4*)h)[i];
  float4 xv = ((const float4*)xw)[i];
  float4 bv = ((const float4*)b)[f4];
  float4 r;
  r.x = fmaxf(hv.x + xv.x * wsl + bv.x, 0.0f);
  r.y = fmaxf(hv.y + xv.y * wsl + bv.y, 0.0f);
  r.z = fmaxf(hv.z + xv.z * wsl + bv.z, 0.0f);
  r.w = fmaxf(hv.w + xv.w * wsl + bv.w, 0.0f);
  ((float4*)h)[i] = r;
}

// ---------------------------------------------------------------------------
// GEMM2: Z = H[ n x 128 ] * W_out[ 128 x 32 ] + b_out  (fp32 WMMA 16x16x4)
// Block = 64 threads = 2 waves; wave w does the 16x16 tile at columns 16w.
// ---------------------------------------------------------------------------
__global__ void __launch_bounds__(64) gemm2_wmma(const float* __restrict__ H,
                                                 const float* __restrict__ W,
                                                 const float* __restrict__ bias,
                                                 float* __restrict__ Z,
                                                 int n_nodes) {
  const int tid = threadIdx.x;
  const int rowBase = blockIdx.x * 16;
  const int wave = tid >> 5;  // 0..1
  const int lane = tid & 31;
  const int half = lane >> 4;
  const int lrow = lane & 15;
  const int colBase = wave * 16;

  v8f acc = {};
  for (int k0 = 0; k0 < 128; k0 += 4) {
    const int ka = k0 + 2 * half;
    v2f a, b;
    const float* ap = H + (long long)(rowBase + lrow) * 128 + ka;
    a[0] = ap[0];
    a[1] = ap[1];
    b[0] = W[ka * 32 + colBase + lrow];
    b[1] = W[(ka + 1) * 32 + colBase + lrow];
    acc = __builtin_amdgcn_wmma_f32_16x16x4_f32(false, a, false, b,
                                                (short)0, acc, false, false);
  }

  float bo = bias[colBase + lrow];
#pragma unroll
  for (int v = 0; v < 8; ++v) {
    Z[(long long)(rowBase + v + 8 * half) * 32 + colBase + lrow] = acc[v] + bo;
  }
}

// ---------------------------------------------------------------------------
extern "C" void kernel_launch(void* const* d_in, const int* in_sizes, int n_in,
                              void* d_out, int out_size, void* d_ws, size_t ws_size,
                              hipStream_t stream) {
  const float* x     = (const float*)d_in[0];  // [n, 256]
  const int*   ei    = (const int*)d_in[1];    // [2, E] flat
  const float* W_gcn = (const float*)d_in[2];  // [256, 128]
  const float* b_gcn = (const float*)d_in[3];  // [128]
  const float* W_out = (const float*)d_in[4];  // [128, 32]
  const float* b_out = (const float*)d_in[5];  // [32]

  const int n_nodes = in_sizes[0] / 256;
  const int E = in_sizes[1] / 2;

  float* h = (float*)d_out;                             // [n, 128] output 0
  float* z = (float*)d_out + (long long)n_nodes * 128;  // [n, 32]  output 1

  float* xw  = (float*)d_ws;                            // [n, 128] scratch
  float* deg = xw + (long long)n_nodes * 128;           // [n] deg -> dinv

  const long long totalH = (long long)n_nodes * 128;

  init_kernel<<<(int)((totalH + 255) / 256), 256, 0, stream>>>(h, deg, n_nodes);
  deg_kernel<<<(E + 255) / 256, 256, 0, stream>>>(ei + E, deg, E);
  rsqrt_kernel<<<(n_nodes + 255) / 256, 256, 0, stream>>>(deg, n_nodes);

  gemm1_wmma<<<n_nodes / 16, 256, 0, stream>>>(x, W_gcn, xw, n_nodes);

  scatter_kernel<<<(E + 7) / 8, 256, 0, stream>>>(ei, xw, deg, h, E);

  const long long q = (long long)n_nodes * 32;
  finalize_kernel<<<(int)((q + 255) / 256), 256, 0, stream>>>(h, xw, deg, b_gcn, n_nodes);

  gemm2_wmma<<<n_nodes / 16, 64, 0, stream>>>(h, W_out, b_out, z, n_nodes);
}